// StatMem_9225589752446
// MI455X (gfx1250) — compile-verified
//
#include <hip/hip_runtime.h>

// StatMem fused window-conv (WMMA f32 16x16x4) + first-order scan.
// B=8, T=2048, M=8, D=256, WINDOW=8, ALPHA=0.5.
// One wave per (b, m, 16-wide d-tile) chain: 1024 waves, 128 t-tiles each.
// Memory-bound: 268 MB @ 23.3 TB/s ~= 11.5 us floor; z read once (NT load),
// h written once (NT store), window overlap carried in B-operand VGPRs.

typedef float v2f __attribute__((ext_vector_type(2)));
typedef float v8f __attribute__((ext_vector_type(8)));

#if __has_builtin(__builtin_amdgcn_wmma_f32_16x16x4_f32)
#define HAVE_WMMA_F32X4 1
#endif

#if __has_builtin(__builtin_amdgcn_rcpf)
__device__ __forceinline__ float fast_rcp(float x) { return __builtin_amdgcn_rcpf(x); }
#else
__device__ __forceinline__ float fast_rcp(float x) { return 1.0f / x; }
#endif

#define B_ 8
#define T_ 2048
#define M_ 8
#define D_ 256
#define MD_ (M_ * D_)

// General window weight (only used to build the first-tile LDS table).
__device__ __forceinline__ float wcoef_slow(int t, int k) {
  int L = (t + 1 < 8) ? (t + 1) : 8;
  bool valid = (k >= 0) & (k < L);
  float raw = (float)(L - 1 - 2 * k);
  float nrm = (float)((L * L - (L & 1)) >> 1);  // L even: L^2/2, odd: (L^2-1)/2
  float inv = fast_rcp(fmaxf(nrm, 1.0f));       // L==1: raw==0 anyway
  return valid ? raw * inv : 0.0f;
}

// A-matrix element: banded weight * v.  FIRST reads the per-row LDS table.
// Branchless: clamp the table index (always in-bounds), select on the value.
template <bool FIRST>
__device__ __forceinline__ float aval(const float (*wt)[8], int i, int j, float vj) {
  int k = i - j + 8;  // window tap index
  bool inband = ((unsigned)k <= 7u);
  float w;
  if (FIRST) {
    float wl = wt[i][k & 7];  // unconditional ds_load, address always valid
    w = inband ? wl : 0.0f;
  } else {
    w = inband ? (float)(7 - 2 * k) * 0.03125f : 0.0f;
  }
  return w * vj;
}

template <bool FIRST>
__device__ __forceinline__ void process_tile(
    int t0, const float* __restrict__ zp, const float* __restrict__ vp,
    float* __restrict__ hp, int lane, int half, int n,
    float (&vl)[32], const float (*wt)[8], v2f (&bC)[6], float& hprev) {
  // ---- stage v window v[t0-8 .. t0+15] into wave-local LDS (DS in-order) ----
  if (lane < 24) {
    int r = t0 - 8 + lane;
    int rc = (FIRST && r < 0) ? 0 : r;        // clamp address, select value
    float v = vp[(size_t)rc * M_];
    if (FIRST) v = (r < 0) ? 0.0f : v;
    vl[lane] = v;
  }

  // ---- prefetch next tile's 16 z rows (2 rows per instruction via halves) --
  if (!FIRST && t0 + 16 < T_) {
#pragma unroll
    for (int c = 0; c < 8; ++c) {
      // half 0 -> even rows t0+16..t0+30, half 1 -> odd rows t0+17..t0+31
      __builtin_prefetch(zp + (size_t)(t0 + 16 + half + 2 * c) * MD_, 0, 0);
    }
  }

  // ---- load B chunks 2..5 (new z rows t0 .. t0+15); 0..1 carried over ----
  // B 4x16 layout per chunk: vgpr0 = row 4kc+2*half, vgpr1 = row 4kc+2*half+1
#pragma unroll
  for (int kc = 2; kc < 6; ++kc) {
    int r = t0 - 8 + 4 * kc + 2 * half;
    bC[kc].x = __builtin_nontemporal_load(&zp[(size_t)r * MD_]);
    bC[kc].y = __builtin_nontemporal_load(&zp[(size_t)(r + 1) * MD_]);
  }

  // ---- build banded A (16x24) per 4-column chunk and chain 6 WMMAs ----
  v8f c = {0.f, 0.f, 0.f, 0.f, 0.f, 0.f, 0.f, 0.f};
#pragma unroll
  for (int kc = 0; kc < 6; ++kc) {
    int j0 = 4 * kc + 2 * half;  // A layout: vgpr0 K=j0, vgpr1 K=j0+1
    float vj0 = vl[j0];
    float vj1 = vl[j0 + 1];
    v2f a;
    a.x = aval<FIRST>(wt, n, j0, vj0);  // A row index = lane & 15
    a.y = aval<FIRST>(wt, n, j0 + 1, vj1);
#ifdef HAVE_WMMA_F32X4
    c = __builtin_amdgcn_wmma_f32_16x16x4_f32(
        /*neg_a=*/false, a, /*neg_b=*/false, bC[kc],
        /*c_mod=*/(short)0, c, /*reuse_a=*/false, /*reuse_b=*/false);
#else
    c[0] += a.x * bC[kc].x + a.y * bC[kc].y;  // placeholder; flags missing builtin
#endif
  }

  // ---- per-row scalars: v window into registers (const-index after unroll) --
  int i0 = 8 * half;  // C layout: lane<16 rows 0-7, else 8-15
  float vr[15];
#pragma unroll
  for (int q = 0; q < 15; ++q) vr[q] = vl[i0 + 1 + q];  // v[t0+i0-7 .. t0+i0+7]

  // ---- segmented scan over the tile's 16 rows ------------------------------
  float out_[8], pref[8];
  float hacc = half ? 0.0f : hprev;
  float p = 1.0f;
#pragma unroll
  for (int r = 0; r < 8; ++r) {
    float den;
    if (FIRST) {
      den = 0.0f;
#pragma unroll
      for (int k = 0; k < 8; ++k)
        den += fabsf(wt[i0 + r][k]) * vr[r + 7 - k];  // consecutive k -> b128
    } else {
      den = 0.21875f * vr[r + 7] + 0.15625f * vr[r + 6] +
            0.09375f * vr[r + 5] + 0.03125f * vr[r + 4] +
            0.03125f * vr[r + 3] + 0.09375f * vr[r + 2] +
            0.15625f * vr[r + 1] + 0.21875f * vr[r + 0];
    }
    float vt = vr[r + 7];  // v[t]
    float inv = fast_rcp(fmaxf(den, 1e-6f));
    float bt = (0.5f * vt) * (c[r] * inv);  // alpha*v*arp
    float at = fmaf(-0.5f, vt, 1.0f);       // 1 - alpha*v
    hacc = fmaf(at, hacc, bt);
    p *= at;
    out_[r] = hacc;
    pref[r] = p;
  }
  // carry h across the lane halves: upper half adds pref[r]*h7
  float h7 = __shfl_xor(out_[7], 16, 32);
  float corr = half ? h7 : 0.0f;
#pragma unroll
  for (int r = 0; r < 8; ++r) out_[r] = fmaf(pref[r], corr, out_[r]);
  float o7 = out_[7];
  float ox = __shfl_xor(o7, 16, 32);
  hprev = half ? o7 : ox;  // h[t0+15] for all lanes

  // ---- store 16 h rows (streaming, written exactly once) -------------------
#pragma unroll
  for (int r = 0; r < 8; ++r)
    __builtin_nontemporal_store(out_[r], &hp[(size_t)(t0 + i0 + r) * MD_]);

  // ---- carry z-row overlap into next tile's chunks 0,1 ---------------------
  bC[0] = bC[4];
  bC[1] = bC[5];
}

__global__ __launch_bounds__(256, 1) void statmem_wmma(
    const float* __restrict__ z, const float* __restrict__ vm,
    float* __restrict__ out) {
  __shared__ float vlds[8][32];
  __shared__ float wtab[8][16][8];  // first-tile ramp weights, per wave slot
  int lane = threadIdx.x & 31;
  int wslot = threadIdx.x >> 5;
  int wid = blockIdx.x * 8 + wslot;  // 0..1023
  int half = lane >> 4;
  int n = lane & 15;
  int dt = wid & 15;  // d-tile
  int m = (wid >> 4) & 7;
  int b = wid >> 7;
  int d = dt * 16 + n;

  const float* zp = z + ((size_t)(b * T_) * M_ + m) * D_ + d;  // + r*MD_
  const float* vp = vm + (size_t)(b * T_) * M_ + m;            // + r*M_
  float* hp = out + ((size_t)(b * T_) * M_ + m) * D_ + d;      // + t*MD_
  float* memf = out + (size_t)B_ * T_ * M_ * D_ + (size_t)(b * M_ + m) * D_ + d;

  // Build first-tile weight table: lanes 0..15 each fill row t = lane.
  if (lane < 16) {
#pragma unroll
    for (int k = 0; k < 8; ++k) wtab[wslot][lane][k] = wcoef_slow(lane, k);
  }

  v2f bC[6];
#pragma unroll
  for (int q = 0; q < 6; ++q) { bC[q].x = 0.0f; bC[q].y = 0.0f; }
  float hprev = 0.0f;

  process_tile<true>(0, zp, vp, hp, lane, half, n, vlds[wslot], wtab[wslot], bC, hprev);
  for (int t0 = 16; t0 < T_; t0 += 16) {
    process_tile<false>(t0, zp, vp, hp, lane, half, n, vlds[wslot], wtab[wslot], bC, hprev);
  }
  if (half == 0) *memf = hprev;  // mem_final = h[:, T-1]
}

extern "C" void kernel_launch(void* const* d_in, const int* in_sizes, int n_in,
                              void* d_out, int out_size, void* d_ws, size_t ws_size,
                              hipStream_t stream) {
  const float* z = (const float*)d_in[0];   // (B,T,M,D) fp32
  const float* vm = (const float*)d_in[1];  // (B,T,M)   fp32
  float* out = (float*)d_out;               // h (B,T,M,D) ++ mem_final (B,M,D)
  (void)in_sizes; (void)n_in; (void)out_size; (void)d_ws; (void)ws_size;
  // 1024 waves total = 128 blocks * 8 waves
  statmem_wmma<<<dim3(128), dim3(256), 0, stream>>>(z, vm, out);
}